// SimpleBinomialCrossEntropyWithLogits_11716670783833
// MI455X (gfx1250) — compile-verified
//
#include <hip/hip_runtime.h>
#include <math.h>

#define DEV __device__ __forceinline__

static constexpr int NXD   = 64;
static constexpr int NVOX  = NXD * NXD * NXD;   // 262144
static constexpr int TW    = 10;                // 8 + 2 halo
static constexpr int TVOL  = TW * TW * TW;      // 1000

// ---------------------------------------------------------------------------
// Compile-time 27-cell adjacency bitmasks (18-adjacency / 26-adjacency),
// index j = (dz+1)*9 + (dy+1)*3 + (dx+1) exactly like the reference _POS.
// ---------------------------------------------------------------------------
struct AdjT { unsigned a18[27]; unsigned a26[27]; };

constexpr AdjT build_adj() {
  AdjT t{};
  for (int i = 0; i < 27; ++i) {
    int iz = i / 9, iy = (i / 3) % 3, ix = i % 3;
    unsigned m18 = 0, m26 = 0;
    for (int j = 0; j < 27; ++j) {
      int jz = j / 9, jy = (j / 3) % 3, jx = j % 3;
      int dz = iz > jz ? iz - jz : jz - iz;
      int dy = iy > jy ? iy - jy : jy - iy;
      int dx = ix > jx ? ix - jx : jx - ix;
      if (dz + dy + dx <= 2) m18 |= 1u << j;
      if (dz <= 1 && dy <= 1 && dx <= 1) m26 |= 1u << j;
    }
    t.a18[i] = m18; t.a26[i] = m26;
  }
  return t;
}

__device__ constexpr AdjT kAdj = build_adj();

// ---------------------------------------------------------------------------
// Component count of a 27-bit mask under adjacency masks: bitwise flood fill.
// Equivalent to the reference's 26-step min-label propagation count.
// ---------------------------------------------------------------------------
DEV int count_comp27(unsigned mask, const unsigned* __restrict__ adj) {
  int cnt = 0;
  while (mask) {
    unsigned comp = mask & (0u - mask);      // seed = lowest set bit
    for (;;) {
      unsigned grow = comp, f = comp;
      while (f) {                            // dilate by adjacency
        unsigned b = (unsigned)__builtin_ctz(f);
        f &= f - 1u;
        grow |= adj[b];
      }
      grow &= mask;
      if (grow == comp) break;
      comp = grow;
    }
    mask &= ~comp;
    ++cnt;
  }
  return cnt;
}

// ---------------------------------------------------------------------------
// Union-find (atomicMin linking, min-index roots == reference's min-label CC)
// ---------------------------------------------------------------------------
DEV int find_root(const int* L, int i) {
  int p = L[i];
  while (p != i) { i = p; p = L[i]; }
  return i;
}

DEV void unite(int* L, int a, int b) {
  for (;;) {
    a = find_root(L, a);
    b = find_root(L, b);
    if (a == b) return;
    int hi = a > b ? a : b;
    int lo = a > b ? b : a;
    int old = atomicMin(&L[hi], lo);
    if (old == hi) return;
    a = lo; b = old;
  }
}

// ---------------------------------------------------------------------------
// Kernel 1: init labels (identity) for both volumes
// ---------------------------------------------------------------------------
__global__ void k_init(int* __restrict__ Lp, int* __restrict__ Lt) {
  int i = blockIdx.x * blockDim.x + threadIdx.x;
  if (i < NVOX) { Lp[i] = i; Lt[i] = i; }
}

// ---------------------------------------------------------------------------
// Kernel 2: merge 6-neighbors among foreground (x > 0.5). One pass suffices.
// First NVOX threads handle pred/Lp, next NVOX handle target/Lt.
// ---------------------------------------------------------------------------
__global__ void k_merge(const float* __restrict__ pred, const float* __restrict__ tgt,
                        int* __restrict__ Lp, int* __restrict__ Lt) {
  int g = blockIdx.x * blockDim.x + threadIdx.x;
  const float* x = (g < NVOX) ? pred : tgt;
  int*         L = (g < NVOX) ? Lp   : Lt;
  int i = (g < NVOX) ? g : g - NVOX;
  if (i >= NVOX) return;
  if (!(x[i] > 0.5f)) return;
  int z = i >> 12, y = (i >> 6) & 63, xc = i & 63;
  if (xc < 63 && x[i + 1]    > 0.5f) unite(L, i, i + 1);
  if (y  < 63 && x[i + 64]   > 0.5f) unite(L, i, i + 64);
  if (z  < 63 && x[i + 4096] > 0.5f) unite(L, i, i + 4096);
}

// ---------------------------------------------------------------------------
// Kernel 3: flatten to final root labels
// ---------------------------------------------------------------------------
__global__ void k_flatten(int* __restrict__ Lp, int* __restrict__ Lt) {
  int g = blockIdx.x * blockDim.x + threadIdx.x;
  int* L = (g < NVOX) ? Lp : Lt;
  int i = (g < NVOX) ? g : g - NVOX;
  if (i >= NVOX) return;
  L[i] = find_root(L, i);
}

// ---------------------------------------------------------------------------
// Loss helpers
// ---------------------------------------------------------------------------
DEV float softplusf(float x) {
  return log1pf(expf(-fabsf(x))) + fmaxf(x, 0.0f);
}

DEV int clamp63(int v) { return v < 0 ? 0 : (v > 63 ? 63 : v); }

// weight*x for one voxel from its LDS label tile (10x10x10, center at +1)
DEV float weighted_val(const int* __restrict__ tile, int lx, int ly, int lz, float v) {
  int ci = ((lz + 1) * TW + (ly + 1)) * TW + (lx + 1);
  int c = tile[ci];
  unsigned eq = 0;
#pragma unroll
  for (int j = 0; j < 27; ++j) {
    int dz = j / 9 - 1, dy = (j / 3) % 3 - 1, dx = j % 3 - 1;
    int lab = tile[ci + dz * (TW * TW) + dy * TW + dx];
    eq |= (unsigned)(lab == c) << j;
  }
  unsigned neq = (~eq) & 0x7FFFFFFu;           // nb != center (center excluded)
  unsigned m2  = eq & ~(1u << 13);             // nb == center, minus center
  int c1 = count_comp27(neq, kAdj.a18);
  int c2 = count_comp27(m2,  kAdj.a26);
  bool fg = v > 0.5f;
  bool ns = fg && ((c1 != 1) || (c2 != 1));
  return (ns ? 10.0f : 1.0f) * v;
}

// ---------------------------------------------------------------------------
// Wave-level f32 sum via V_WMMA_F32_16X16X4_F32 with B = ones.
// Each lane's value appears exactly once in A; D rows replicate row-sums;
// summing a lane's 8 D regs gives half the rows, shfl_xor(16) adds the rest.
// ---------------------------------------------------------------------------
typedef float v2f_t __attribute__((ext_vector_type(2)));
typedef float v8f_t __attribute__((ext_vector_type(8)));

DEV float wave_sum32(float v) {
#if __has_builtin(__builtin_amdgcn_wmma_f32_16x16x4_f32)
  v2f_t a; a[0] = v;    a[1] = 0.0f;
  v2f_t b; b[0] = 1.0f; b[1] = 1.0f;
  v8f_t c = {};
  c = __builtin_amdgcn_wmma_f32_16x16x4_f32(false, a, false, b, (short)0, c, false, false);
  float s = c[0] + c[1] + c[2] + c[3] + c[4] + c[5] + c[6] + c[7];
  s += __shfl_xor(s, 16, 32);
  return s;
#else
#warning "CDNA5: wmma_f32_16x16x4_f32 builtin unavailable; falling back to shuffles"
  for (int m = 16; m > 0; m >>= 1) v += __shfl_xor(v, m, 32);
  return v;
#endif
}

// ---------------------------------------------------------------------------
// Kernel 4: per-voxel non-simple weights + BCE-with-logits, block partials.
// 8x8x8 tile per block; halo label tiles staged via async global->LDS.
// ---------------------------------------------------------------------------
typedef __attribute__((address_space(1))) int g_i32;   // global (AS1) int
typedef __attribute__((address_space(3))) int l_i32;   // LDS (AS3) int

__global__ void __launch_bounds__(256)
k_loss(const float* __restrict__ pred, const float* __restrict__ tgt,
       const int* __restrict__ Lp, const int* __restrict__ Lt,
       float* __restrict__ bsum) {
  __shared__ int tp[TVOL];
  __shared__ int tt[TVOL];
  __shared__ float wpart[8];

  const int tid = threadIdx.x;
  const int bx = blockIdx.x * 8, by = blockIdx.y * 8, bz = blockIdx.z * 8;

  // Stage 2 x 1000 halo labels (edge-clamped) into LDS asynchronously.
  for (int h = tid; h < 2 * TVOL; h += 256) {
    int pos = (h < TVOL) ? h : h - TVOL;
    int hz = pos / (TW * TW), hy = (pos / TW) % TW, hx = pos % TW;
    int gz = clamp63(bz + hz - 1), gy = clamp63(by + hy - 1), gx = clamp63(bx + hx - 1);
    int gi = (gz << 12) | (gy << 6) | gx;
    const int* src = (h < TVOL) ? Lp : Lt;
    int*       dst = (h < TVOL) ? tp : tt;
#if __has_builtin(__builtin_amdgcn_global_load_async_to_lds_b32)
    __builtin_amdgcn_global_load_async_to_lds_b32(
        (g_i32*)(src + gi), (l_i32*)(dst + pos), 0, 0);
#else
#warning "CDNA5: global_load_async_to_lds_b32 builtin unavailable; synchronous fallback"
    dst[pos] = src[gi];
#endif
  }
#if __has_builtin(__builtin_amdgcn_s_wait_asynccnt)
  __builtin_amdgcn_s_wait_asynccnt(0);
#else
  asm volatile("s_wait_asynccnt 0" ::: "memory");
#endif
  __syncthreads();

  const int lx = tid & 7, ly = (tid >> 3) & 7, lz0 = tid >> 6;
  float acc = 0.0f;
#pragma unroll
  for (int s = 0; s < 2; ++s) {
    int lz = lz0 + 4 * s;
    int gi = ((bz + lz) << 12) | ((by + ly) << 6) | (bx + lx);
    float pv = pred[gi];
    float tv = tgt[gi];
    float wp = weighted_val(tp, lx, ly, lz, pv);
    float wl = weighted_val(tt, lx, ly, lz, tv);
    acc += wl * softplusf(-wp) + (1.0f - wl) * softplusf(wp);
  }

  // Uniform control flow here: EXEC is all-ones for the WMMA reduction.
  float wsum = wave_sum32(acc);
  int lane = tid & 31, wave = tid >> 5;
  if (lane == 0) wpart[wave] = wsum;
  __syncthreads();
  if (tid == 0) {
    float b = 0.0f;
#pragma unroll
    for (int w = 0; w < 8; ++w) b += wpart[w];
    bsum[((blockIdx.z * 8) + blockIdx.y) * 8 + blockIdx.x] = b;
  }
}

// ---------------------------------------------------------------------------
// Kernel 5: deterministic final reduction of 512 block partials.
// ---------------------------------------------------------------------------
__global__ void __launch_bounds__(256)
k_final(const float* __restrict__ bsum, float* __restrict__ out) {
  __shared__ float red[256];
  int tid = threadIdx.x;
  red[tid] = bsum[tid] + bsum[tid + 256];
  __syncthreads();
  for (int s = 128; s > 0; s >>= 1) {
    if (tid < s) red[tid] += red[tid + s];
    __syncthreads();
  }
  if (tid == 0) out[0] = red[0];
}

// ---------------------------------------------------------------------------
extern "C" void kernel_launch(void* const* d_in, const int* in_sizes, int n_in,
                              void* d_out, int out_size, void* d_ws, size_t ws_size,
                              hipStream_t stream) {
  const float* pred = (const float*)d_in[0];
  const float* tgt  = (const float*)d_in[1];
  float* out = (float*)d_out;

  int*   Lp   = (int*)d_ws;
  int*   Lt   = Lp + NVOX;
  float* bsum = (float*)(Lt + NVOX);

  k_init   <<<dim3((NVOX + 255) / 256),     dim3(256), 0, stream>>>(Lp, Lt);
  k_merge  <<<dim3((2 * NVOX + 255) / 256), dim3(256), 0, stream>>>(pred, tgt, Lp, Lt);
  k_flatten<<<dim3((2 * NVOX + 255) / 256), dim3(256), 0, stream>>>(Lp, Lt);
  k_loss   <<<dim3(8, 8, 8),                dim3(256), 0, stream>>>(pred, tgt, Lp, Lt, bsum);
  k_final  <<<dim3(1),                      dim3(256), 0, stream>>>(bsum, out);
}